// GINEncoder_84894323572906
// MI455X (gfx1250) — compile-verified
//
#include <hip/hip_runtime.h>

#define N_NODES  50000
#define IN_DIM   128
#define HID      64
#define OUT_DIM  128
#define N_GRAPHS 128

typedef float v2f __attribute__((ext_vector_type(2)));
typedef float v8f __attribute__((ext_vector_type(8)));

// ---------------------------------------------------------------------------
// WMMA GEMM: C[M,N] = act( (A [+ A2]) @ W + bias )
//   A:[M,K] row-major, W:[K,N] row-major.
//
// V_WMMA_F32_16X16X4_F32, wave32. Each wave owns MT row tiles x N/16 col
// tiles. W is staged ONCE per block into LDS, pre-packed into the per-lane
// B-fragment layout, so the hot loop reads one ds_load_b64 per fragment
// (32 lanes x 8B contiguous -> conflict-free across 64 banks) instead of
// two global_load_b32.
//
// f32 16x16x4 VGPR layout (ISA 7.12.2):
//   A (16x4): lanes 0-15 -> rows, K={0,1}; lanes 16-31 -> rows, K={2,3}.
//   B (4x16): lanes 0-15 -> col=lane, K={0,1}; lanes 16-31 -> K={2,3}.
//   C/D:      vgpr j -> row j (lanes 0-15) / row j+8 (lanes 16-31), col=lane&15.
//
// M % 16 == 0 (50000 = 3125*16). Tail tiles are clamped to the last tile:
// duplicate waves recompute and store identical values (deterministic, no
// OOB access), and EXEC stays all-ones for every WMMA.
// ---------------------------------------------------------------------------
template <int K, int N, int MT, bool RELU, bool HASA2>
__global__ __launch_bounds__(256) void gemm_wmma_kernel(
    const float* __restrict__ A, const float* __restrict__ A2,
    const float* __restrict__ W, const float* __restrict__ bias,
    float* __restrict__ C, int M)
{
  constexpr int NT    = N / 16;
  constexpr int NFRAG = (K / 4) * NT;        // B fragments per GEMM
  __shared__ float2 ldsW[NFRAG * 32];        // <= 32 KB

  const int tid = threadIdx.x;

  // ---- cooperative pack of W into B-fragment layout (all 8 waves) ----
  for (int i = tid; i < NFRAG * 32; i += 256) {
    const int lane_i = i & 31;
    const int frag   = i >> 5;
    const int t      = frag % NT;
    const int kstep  = frag / NT;
    const int kpair  = kstep * 4 + (lane_i >> 4) * 2;
    const int col    = t * 16 + (lane_i & 15);
    ldsW[i] = make_float2(W[(size_t)kpair * N + col],
                          W[(size_t)(kpair + 1) * N + col]);
  }
  __syncthreads();

  const int lane     = tid & 31;
  const int wave     = tid >> 5;
  const int lastTile = M / 16 - 1;

  int mt[MT];
  mt[0] = min(blockIdx.x * (8 * MT) + wave * MT, lastTile);
#pragma unroll
  for (int m = 1; m < MT; ++m) mt[m] = min(mt[m - 1] + 1, lastTile);

  const int l15 = lane & 15;
  const int kh  = lane >> 4;                 // 0 -> K pair {0,1}, 1 -> {2,3}

  const float* Ar[MT];
  const float* A2r[MT];
#pragma unroll
  for (int m = 0; m < MT; ++m) {
    Ar[m]  = A + (size_t)(mt[m] * 16 + l15) * K + 2 * kh;
    A2r[m] = HASA2 ? (A2 + (size_t)(mt[m] * 16 + l15) * K + 2 * kh) : nullptr;
  }

  v8f acc[MT][NT] = {};

#pragma unroll 4
  for (int k0 = 0; k0 < K; k0 += 4) {
    v2f a[MT];
#pragma unroll
    for (int m = 0; m < MT; ++m) {
      a[m].x = Ar[m][k0];
      a[m].y = Ar[m][k0 + 1];
      if (HASA2) { a[m].x += A2r[m][k0]; a[m].y += A2r[m][k0 + 1]; }  // GIN skip
    }
    const float2* bp = &ldsW[(k0 >> 2) * NT * 32 + lane];
#pragma unroll
    for (int t = 0; t < NT; ++t) {
      const float2 b2 = bp[t * 32];
      v2f bf; bf.x = b2.x; bf.y = b2.y;
#pragma unroll
      for (int m = 0; m < MT; ++m) {
        acc[m][t] = __builtin_amdgcn_wmma_f32_16x16x4_f32(
            /*neg_a=*/false, a[m], /*neg_b=*/false, bf,
            /*c_mod=*/(short)0, acc[m][t], /*reuse_a=*/false, /*reuse_b=*/false);
      }
    }
  }

#pragma unroll
  for (int m = 0; m < MT; ++m) {
    const int row0 = mt[m] * 16;
#pragma unroll
    for (int t = 0; t < NT; ++t) {
      const int col = t * 16 + l15;
      const float bv = bias[col];
#pragma unroll
      for (int j = 0; j < 8; ++j) {
        float v = acc[m][t][j] + bv;
        if (RELU) v = fmaxf(v, 0.0f);
        C[(size_t)(row0 + kh * 8 + j) * N + col] = v;
      }
    }
  }
}

// ---------------------------------------------------------------------------
// Edge scatter-add: aggr[dst[e], f] += h[src[e], f]   (64 threads per edge)
// unsafeAtomicAdd -> global_atomic_add_f32 (no CAS loop); targets L2-resident.
// ---------------------------------------------------------------------------
__global__ __launch_bounds__(256) void scatter_add_kernel(
    const float* __restrict__ h, const int* __restrict__ src,
    const int* __restrict__ dst, float* __restrict__ aggr, int E)
{
  const long long t = (long long)blockIdx.x * 256 + threadIdx.x;
  if (t >= (long long)E * HID) return;
  const int e = (int)(t >> 6);
  const int f = (int)(t & (HID - 1));
  const int s = src[e];
  const int d = dst[e];
  unsafeAtomicAdd(&aggr[(size_t)d * HID + f], h[(size_t)s * HID + f]);
}

// ---------------------------------------------------------------------------
__global__ __launch_bounds__(256) void zero_kernel(float4* __restrict__ p, int n4)
{
  const int i = blockIdx.x * 256 + threadIdx.x;
  if (i < n4) p[i] = make_float4(0.f, 0.f, 0.f, 0.f);
}

// Mean-pool stage 1: atomic sums per graph + node counts (f==0 lane counts).
__global__ __launch_bounds__(256) void pool_sum_kernel(
    const float* __restrict__ hn, const int* __restrict__ batch,
    float* __restrict__ sums, float* __restrict__ counts)
{
  const int t = blockIdx.x * 256 + threadIdx.x;
  if (t >= N_NODES * OUT_DIM) return;
  const int node = t >> 7;                   // OUT_DIM == 128
  const int f    = t & (OUT_DIM - 1);
  const int g    = batch[node];
  unsafeAtomicAdd(&sums[(size_t)g * OUT_DIM + f], hn[t]);
  if (f == 0) unsafeAtomicAdd(&counts[g], 1.0f);
}

// Mean-pool stage 2: in-place divide by max(count, 1).
__global__ __launch_bounds__(256) void pool_div_kernel(
    float* __restrict__ out, const float* __restrict__ counts)
{
  const int t = blockIdx.x * 256 + threadIdx.x;
  if (t < N_GRAPHS * OUT_DIM) out[t] /= fmaxf(counts[t >> 7], 1.0f);
}

// ---------------------------------------------------------------------------
extern "C" void kernel_launch(void* const* d_in, const int* in_sizes, int n_in,
                              void* d_out, int out_size, void* d_ws, size_t ws_size,
                              hipStream_t stream)
{
  const float* x     = (const float*)d_in[0];
  const int*   ei    = (const int*)d_in[1];
  const int*   batch = (const int*)d_in[2];
  const float* W_in  = (const float*)d_in[3];
  const float* b_in  = (const float*)d_in[4];
  const float* W1_0  = (const float*)d_in[5];
  const float* b1_0  = (const float*)d_in[6];
  const float* W2_0  = (const float*)d_in[7];
  const float* b2_0  = (const float*)d_in[8];
  const float* W1_1  = (const float*)d_in[9];
  const float* b1_1  = (const float*)d_in[10];
  const float* W2_1  = (const float*)d_in[11];
  const float* b2_1  = (const float*)d_in[12];
  const float* W_out = (const float*)d_in[13];
  const float* b_out = (const float*)d_in[14];
  float* out = (float*)d_out;

  const int E = in_sizes[1] / 2;             // edge_index is [2, E] flat
  const int* src = ei;
  const int* dst = ei + E;

  // Workspace layout (bytes):
  const size_t SZ_H = (size_t)N_NODES * HID * sizeof(float);       // 12.8 MB
  char* ws = (char*)d_ws;
  float* buf0   = (float*)(ws);                                    // h      [N, HID]
  float* buf1   = (float*)(ws + SZ_H);                             // aggr   [N, HID]
  float* buf2   = (float*)(ws + 2 * SZ_H);                         // mid    [N, HID]
  float* buf3   = (float*)(ws + 3 * SZ_H);                         // nodes  [N, OUT]
  float* counts = (float*)(ws + 3 * SZ_H +
                           (size_t)N_NODES * OUT_DIM * sizeof(float));

  const int nTiles   = N_NODES / 16;                               // 3125
  const int grid_mt2 = (nTiles + 15) / 16;                         // 196
  const int grid_mt1 = (nTiles + 7) / 8;                           // 391
  const int zeroH    = (N_NODES * HID / 4 + 255) / 256;
  const unsigned scatterGrid =
      (unsigned)(((long long)E * HID + 255) / 256);

  // 1) proj_in: h = x @ W_in + b_in
  gemm_wmma_kernel<IN_DIM, HID, 2, false, false>
      <<<grid_mt2, 256, 0, stream>>>(x, nullptr, W_in, b_in, buf0, N_NODES);

  // 2) GIN conv 0
  zero_kernel<<<zeroH, 256, 0, stream>>>((float4*)buf1, N_NODES * HID / 4);
  scatter_add_kernel<<<scatterGrid, 256, 0, stream>>>(buf0, src, dst, buf1, E);
  gemm_wmma_kernel<HID, HID, 2, true, true>
      <<<grid_mt2, 256, 0, stream>>>(buf0, buf1, W1_0, b1_0, buf2, N_NODES);
  gemm_wmma_kernel<HID, HID, 2, true, false>      // outer relu fused
      <<<grid_mt2, 256, 0, stream>>>(buf2, nullptr, W2_0, b2_0, buf0, N_NODES);

  // 3) GIN conv 1
  zero_kernel<<<zeroH, 256, 0, stream>>>((float4*)buf1, N_NODES * HID / 4);
  scatter_add_kernel<<<scatterGrid, 256, 0, stream>>>(buf0, src, dst, buf1, E);
  gemm_wmma_kernel<HID, HID, 2, true, true>
      <<<grid_mt2, 256, 0, stream>>>(buf0, buf1, W1_1, b1_1, buf2, N_NODES);
  gemm_wmma_kernel<HID, HID, 2, true, false>
      <<<grid_mt2, 256, 0, stream>>>(buf2, nullptr, W2_1, b2_1, buf0, N_NODES);

  // 4) proj_out: hn = h @ W_out + b_out
  gemm_wmma_kernel<HID, OUT_DIM, 1, false, false>
      <<<grid_mt1, 256, 0, stream>>>(buf0, nullptr, W_out, b_out, buf3, N_NODES);

  // 5) global mean pool
  zero_kernel<<<(N_GRAPHS * OUT_DIM / 4 + 255) / 256, 256, 0, stream>>>(
      (float4*)out, N_GRAPHS * OUT_DIM / 4);
  zero_kernel<<<1, 256, 0, stream>>>((float4*)counts, N_GRAPHS / 4);
  pool_sum_kernel<<<(N_NODES * OUT_DIM + 255) / 256, 256, 0, stream>>>(
      buf3, batch, out, counts);
  pool_div_kernel<<<(N_GRAPHS * OUT_DIM + 255) / 256, 256, 0, stream>>>(
      out, counts);
}